// BSpline3D_86998857548337
// MI455X (gfx1250) — compile-verified
//
#include <hip/hip_runtime.h>

typedef float v2f __attribute__((ext_vector_type(2)));
typedef float v4f __attribute__((ext_vector_type(4)));
typedef float v8f __attribute__((ext_vector_type(8)));

#define CH 32
#define NK 10
#define DHW (64 * 64 * 64)                       // 262144 spatial elems per (b,c) slice
#define PASSES 8
#define ELEMS_PER_BLOCK (256 * PASSES * 4)       // 8192: 256 threads x 8 x v4f
#define BLOCKS_PER_SLICE (DHW / ELEMS_PER_BLOCK) // 32
#define NREG 10                                  // regions 1..10 (region 0 == 0 handled by x-clamp)

// Piecewise-cubic evaluation of f(x) = sum_k w_k * relu(x - t_k)^3.
// coef4[iu] = (a0,a1,a2,a3) of the cubic valid on knot interval iu (iu = 0..9).
__device__ __forceinline__ float eval_spline(float xi, const v4f* __restrict__ coef4) {
    float xc = fmaxf(xi, -1.0f);                 // region-0 (x < t0): poly_1(-1) == 0 exactly
    float u  = __fmaf_rn(xc, 4.5f, 4.5f);        // (x+1)/h, h = 2/9 (uniform linspace knots)
    int  iu  = (int)u;                           // u >= 0 -> trunc == floor
    iu = iu > (NREG - 1) ? (NREG - 1) : iu;
    v4f cf = coef4[iu];                          // ds_load_b128, broadcast/conflict-free
    return __fmaf_rn(__fmaf_rn(__fmaf_rn(cf.w, xc, cf.z), xc, cf.y), xc, cf.x);
}

// Branch-free Mw[k][n] = w_k * (-t^3, 3t^2, -3t, 1)[n], 0 for n>3 or k>=NK.
__device__ __forceinline__ float moment(const float* __restrict__ w,
                                        const float* __restrict__ knots,
                                        int c, int k, int n) {
    const int kc = k < (NK - 1) ? k : (NK - 1);  // clamp so loads are unconditional
    const float wk = w[c * NK + kc];
    const float t  = knots[kc];
    float v = (n == 0) ? -wk * t * t * t
            : (n == 1) ?  3.f * wk * t * t
            : (n == 2) ? -3.f * wk * t
            : (n == 3) ?  wk
                       :  0.f;                   // pure v_cndmask chain, no EXEC branches
    return (k < NK) ? v : 0.f;
}

__global__ __launch_bounds__(256) void bspline3_pwcubic_kernel(
    const float* __restrict__ x, const float* __restrict__ knots,
    const float* __restrict__ w, float* __restrict__ out) {
    __shared__ v4f coef4[NREG];

    const int tid = threadIdx.x;
    const int c   = (blockIdx.x / BLOCKS_PER_SLICE) & (CH - 1);

    // ---- Wave 0: build the per-channel coefficient table on the tensor core ----
    // Coef[r][p] = sum_{k<r} (w_k * T[k][p]),  T[k] = (-t^3, 3t^2, -3t, 1)
    // = L[16x12] @ Mw[12x4] via 3 chained V_WMMA_F32_16X16X4_F32 (fp32-exact).
    if (tid < 32) {                                  // wave32: exactly wave 0, EXEC all-ones
        const int lane = tid;
        const int m    = lane & 15;                  // A row (region r) / B,D column
        const int half = lane >> 4;

        v8f acc = {0.f, 0.f, 0.f, 0.f, 0.f, 0.f, 0.f, 0.f};
#pragma unroll
        for (int j = 0; j < 3; ++j) {
            const int kx = 4 * j + half * 2;         // A/B VGPR0 holds K = 0 (lanes 0-15) / 2 (16-31)
            const int ky = kx + 1;                   // A/B VGPR1 holds K = 1 / 3

            v2f a, b;
            a.x = (kx < m && kx < NK) ? 1.0f : 0.0f; // prefix-sum indicator L[m][k] = (k < m)
            a.y = (ky < m && ky < NK) ? 1.0f : 0.0f;
            b.x = moment(w, knots, c, kx, m);
            b.y = moment(w, knots, c, ky, m);

            acc = __builtin_amdgcn_wmma_f32_16x16x4_f32(
                /*neg_a=*/false, a, /*neg_b=*/false, b,
                /*c_mod=*/(short)0, acc, /*reuse_a=*/false, /*reuse_b=*/false);
        }

        // D layout (32-bit C/D 16x16): VGPR i, lanes 0-15 -> D[i][lane]; lanes 16-31 -> D[i+8][lane-16].
        // Table row (r-1) <- region r = 1..10, column p = 0..3.
        float* coefs = (float*)coef4;
        if (lane < 4) {                              // p = lane, regions 1..7 from acc[1..7]
#pragma unroll
            for (int r = 1; r <= 7; ++r) coefs[(r - 1) * 4 + lane] = acc[r];
        } else if (lane >= 16 && lane < 20) {        // p = lane-16, regions 8..10 from acc[0..2]
            const int p = lane - 16;
#pragma unroll
            for (int i = 0; i < 3; ++i) coefs[(7 + i) * 4 + p] = acc[i];
        }
    }
    __syncthreads();

    // ---- Streaming phase: HBM-bound b128 NT loads/stores, ~8 VALU + 1 LDS gather per elem ----
    const v4f* __restrict__ x4 = (const v4f*)x;
    v4f* __restrict__ o4       = (v4f*)out;
    const int base4 = blockIdx.x * (ELEMS_PER_BLOCK / 4);

#pragma unroll
    for (int p = 0; p < PASSES; ++p) {
        const int i4 = base4 + p * 256 + tid;        // fully coalesced 16B per lane
        v4f xv = __builtin_nontemporal_load(&x4[i4]);
        v4f r;
        r.x = eval_spline(xv.x, coef4);
        r.y = eval_spline(xv.y, coef4);
        r.z = eval_spline(xv.z, coef4);
        r.w = eval_spline(xv.w, coef4);
        __builtin_nontemporal_store(r, &o4[i4]);
    }
}

extern "C" void kernel_launch(void* const* d_in, const int* in_sizes, int n_in,
                              void* d_out, int out_size, void* d_ws, size_t ws_size,
                              hipStream_t stream) {
    const float* x     = (const float*)d_in[0];   // [2,32,64,64,64] fp32
    const float* knots = (const float*)d_in[1];   // [10] fp32
    const float* w     = (const float*)d_in[2];   // [32,10] fp32
    float* out         = (float*)d_out;           // [2,32,64,64,64] fp32

    const int total  = in_sizes[0];               // 16,777,216
    const int blocks = total / ELEMS_PER_BLOCK;   // 2048
    hipLaunchKernelGGL(bspline3_pwcubic_kernel, dim3(blocks), dim3(256), 0, stream,
                       x, knots, w, out);
}